// CycleConsistencyLoss_16097537425579
// MI455X (gfx1250) — compile-verified
//
#include <hip/hip_runtime.h>
#include <math.h>

// ---------------------------------------------------------------------------
// CycleConsistencyLoss on MI455X (gfx1250)
//
// Shapes: B=8, H=16, NT=512, NV=576, 2 layers, fp32.
// Only the diagonals of (P_t@P_v) / (P_v@P_t) are needed:
//   M[b,i,j]       = S_t[b,i,j] * S_v[b,j,i]     (S = head SUM; /256 later)
//   text_diag[b,i] = sum_j M / 256   (row sums)
//   vis_diag[b,j]  = sum_i M / 256   (col sums)
// Memory-bound: 604 MB read once -> ~26us floor at 23.3 TB/s.
// Row/col sums of each 16x16 M tile are done with V_WMMA_F32_16X16X4_F32
// using an all-ones A operand (D[m,n] = sum_k B[k,n]).
// ---------------------------------------------------------------------------

typedef __attribute__((ext_vector_type(2))) float v2f;
typedef __attribute__((ext_vector_type(8))) float v8f;

#define B_  8
#define H_  16
#define NT_ 512
#define NV_ 576
#define TI_TILES (NT_ / 16)          // 32
#define TJ_TILES (NV_ / 16)          // 36
#define TXT_N (2 * B_ * NT_)         // 8192 text-diag accumulators (2 layers)
#define VIS_N (2 * B_ * NV_)         // 9216 vision-diag accumulators

__global__ __launch_bounds__(256)
void ccl_zero_kernel(float* __restrict__ acc, int n) {
  int i = blockIdx.x * blockDim.x + threadIdx.x;
  if (i < n) acc[i] = 0.0f;
}

__global__ __launch_bounds__(256)
void ccl_diag_kernel(const float* __restrict__ t0, const float* __restrict__ v0,
                     const float* __restrict__ t1, const float* __restrict__ v1,
                     float* __restrict__ txtAcc, float* __restrict__ visAcc) {
  // Per-wave staging tiles; pitch 17 -> conflict-free row AND column access.
  __shared__ float ldsT[8][16][17];   // S_t tile, [i][j]
  __shared__ float ldsV[8][16][17];   // S_v^T tile, [i][j] == S_v[j][i]

  const int lane = threadIdx.x & 31;
  const int wid  = threadIdx.x >> 5;
  const int gw   = blockIdx.x * 8 + wid;          // one 16x16 tile per wave

  const int tj = gw % TJ_TILES;
  int rem      = gw / TJ_TILES;
  const int ti = rem % TI_TILES;  rem /= TI_TILES;
  const int b  = rem % B_;
  const int l  = rem / B_;                        // layer 0/1; grid exact fit

  const float* __restrict__ at = l ? t1 : t0;
  const float* __restrict__ av = l ? v1 : v0;

  const int i0 = ti * 16, j0 = tj * 16;
  const int hi = lane >> 4;                       // half-wave (0/1)
  const int lo = lane & 15;

  // ---- head sums of attn_t: lanes sweep j (contiguous, 64B/half-wave) ----
  #pragma unroll
  for (int t = 0; t < 8; ++t) {
    const int i = 2 * t + hi;
    const float* p = at + ((size_t)b * H_ * NT_ + (size_t)(i0 + i)) * NV_ + (j0 + lo);
    float acc = 0.0f;
    #pragma unroll
    for (int h = 0; h < H_; ++h) acc += p[(size_t)h * NT_ * NV_];
    ldsT[wid][i][lo] = acc;
  }

  // ---- head sums of attn_v: lanes sweep i (contiguous, 64B/half-wave) ----
  #pragma unroll
  for (int t = 0; t < 8; ++t) {
    const int j = 2 * t + hi;
    const float* p = av + ((size_t)b * H_ * NV_ + (size_t)(j0 + j)) * NT_ + (i0 + lo);
    float acc = 0.0f;
    #pragma unroll
    for (int h = 0; h < H_; ++h) acc += p[(size_t)h * NV_ * NT_];
    ldsV[wid][lo][j] = acc;                       // store transposed: [i=lo][j]
  }
  // Same-wave LDS store->load: DS ops are in-order within a wave; no barrier.

  // ---- WMMA reductions: D = Ones(16x4) x B(4x16), accumulated over 4 chunks
  // With all-ones A, D[m,n] = sum over all 16 k of B[k,n]; only the
  // "column = lane mod 16" striping of B matters.
  v2f ones; ones[0] = 1.0f; ones[1] = 1.0f;
  v8f cCol = {};   // column sums of M  -> vision_diag
  v8f cRow = {};   // column sums of M^T -> text_diag (row sums of M)

  #pragma unroll
  for (int c = 0; c < 4; ++c) {
    const int k0 = 4 * c + 2 * hi;
    const int k1 = k0 + 1;
    v2f bcol, brow;
    bcol[0] = ldsT[wid][k0][lo] * ldsV[wid][k0][lo];   // M[k0][lo]
    bcol[1] = ldsT[wid][k1][lo] * ldsV[wid][k1][lo];   // M[k1][lo]
    brow[0] = ldsT[wid][lo][k0] * ldsV[wid][lo][k0];   // M[lo][k0] = M^T[k0][lo]
    brow[1] = ldsT[wid][lo][k1] * ldsV[wid][lo][k1];   // M[lo][k1]
    cCol = __builtin_amdgcn_wmma_f32_16x16x4_f32(
        false, ones, false, bcol, (short)0, cCol, false, false);
    cRow = __builtin_amdgcn_wmma_f32_16x16x4_f32(
        false, ones, false, brow, (short)0, cRow, false, false);
  }

  // C/D layout: VGPR 0, lanes 0..15 hold row M=0, N=lane -> the (replicated)
  // column sum for column `lo` lives in cX[0] of lanes 0..15.
  if (lane < 16) {
    atomicAdd(&visAcc[(size_t)l * (B_ * NV_) + b * NV_ + j0 + lane], cCol[0]);
    atomicAdd(&txtAcc[(size_t)l * (B_ * NT_) + b * NT_ + i0 + lane], cRow[0]);
  }
}

__global__ __launch_bounds__(256)
void ccl_loss_kernel(const float* __restrict__ txtAcc,
                     const float* __restrict__ visAcc,
                     float* __restrict__ out) {
  __shared__ float red[256];
  // total = 0.25 * sum_layers(loss_text + loss_vision)
  const float wt = 0.25f / (float)(B_ * NT_);
  const float wv = 0.25f / (float)(B_ * NV_);
  float s = 0.0f;
  for (int i = threadIdx.x; i < TXT_N; i += 256) {
    float x = txtAcc[i] * (1.0f / 256.0f);              // two head-means
    x = fminf(fmaxf(x, 1e-8f), 1.0f);
    s -= logf(x) * wt;
  }
  for (int i = threadIdx.x; i < VIS_N; i += 256) {
    float x = visAcc[i] * (1.0f / 256.0f);
    x = fminf(fmaxf(x, 1e-8f), 1.0f);
    s -= logf(x) * wv;
  }
  red[threadIdx.x] = s;
  __syncthreads();
  for (int off = 128; off > 0; off >>= 1) {
    if ((int)threadIdx.x < off) red[threadIdx.x] += red[threadIdx.x + off];
    __syncthreads();
  }
  if (threadIdx.x == 0) out[0] = red[0];
}

extern "C" void kernel_launch(void* const* d_in, const int* in_sizes, int n_in,
                              void* d_out, int out_size, void* d_ws, size_t ws_size,
                              hipStream_t stream) {
  (void)in_sizes; (void)n_in; (void)out_size; (void)ws_size;
  const float* t0 = (const float*)d_in[0];
  const float* v0 = (const float*)d_in[1];
  const float* t1 = (const float*)d_in[2];
  const float* v1 = (const float*)d_in[3];

  float* txtAcc = (float*)d_ws;           // [2][B][NT]
  float* visAcc = txtAcc + TXT_N;         // [2][B][NV]
  float* out    = (float*)d_out;

  const int nacc = TXT_N + VIS_N;         // 17408 floats (~68 KB of d_ws)
  ccl_zero_kernel<<<(nacc + 255) / 256, 256, 0, stream>>>(txtAcc, nacc);

  const int total_waves  = 2 * B_ * TI_TILES * TJ_TILES;  // 18432, exact fit
  const int total_blocks = total_waves / 8;               // 2304
  ccl_diag_kernel<<<total_blocks, 256, 0, stream>>>(t0, v0, t1, v1, txtAcc, visAcc);

  ccl_loss_kernel<<<1, 256, 0, stream>>>(txtAcc, visAcc, out);
}